// SubTrajectoryBalanceLoss_64553358459175
// MI455X (gfx1250) — compile-verified
//
#include <hip/hip_runtime.h>

// ---------------------------------------------------------------------------
// Sub-trajectory balance loss, MI455X (gfx1250).
//   d[b,t]     = log_pf[b,t] - log_pb[b,t]
//   residual   = (log_z - log_R[b]) + suffix_sum_t(d)
//   loss       = (1/(B*S)) * sum_{b,t} lambda^(S-1-t) * residual^2
//
// Single-pass, memory-roofline design: 128 MB read exactly once (~5.8us at
// 23.3 TB/s). Each wave owns a 16-row x (S/16)-col chunk and loads it ONCE,
// directly in WMMA A-fragment order, into 128 registers/lane, accumulating
// per-row chunk totals on the fly (lanes m and m+16 jointly hold all of row
// m -> one shfl_xor(16) finishes the row sum). After an LDS carry exchange,
// the suffix scan runs entirely out of registers on the matrix unit:
// Local(16x16) = Dtile x U(lower-tri ones) via 4 chained
// V_WMMA_F32_16X16X4_F32 per tile, carry injected as the WMMA C operand and
// recovered from D column 0 via ds_bpermute. No loads in the scan loop.
// ---------------------------------------------------------------------------

typedef float v2f __attribute__((ext_vector_type(2)));
typedef float v8f __attribute__((ext_vector_type(8)));

#define LAMBDA      0.9f
#define LAMBDA_P16  0.18530201888518410f   // 0.9^16
#define WAVES_PER_WG 16                    // 512 threads, wave32

// ---------------------------------------------------------------------------
// Register-resident single-pass kernel. NT = tiles (of 16 cols) per wave.
// ---------------------------------------------------------------------------
template <int NT>
__global__ void __launch_bounds__(512)
stb_loss_main(const float* __restrict__ pf,
              const float* __restrict__ pb,
              const float* __restrict__ logR,
              const float* __restrict__ logZ,
              float* __restrict__ partials,
              int S) {
    __shared__ float totalsLDS[WAVES_PER_WG][16];
    __shared__ float carryLDS[WAVES_PER_WG][16];
    __shared__ float wsum[WAVES_PER_WG];

    const int tid  = threadIdx.x;
    const int lane = tid & 31;
    const int w    = tid >> 5;            // wave id: owns cols [w*CH, w*CH+CH)
    const int m    = lane & 15;           // row within 16-row block
    const int h    = (lane >> 4) & 1;     // half-wave select
    const int row0 = blockIdx.x * 16;
    const int CH   = NT * 16;
    const int c0   = w * CH;

    // ------ Load whole chunk once, in A-fragment order, into registers -----
    // dv[t][k] = { d[m][c0+16t+4k+2h], d[m][c0+16t+4k+2h+1] }
    const float* pfp = pf + (size_t)(row0 + m) * S + (c0 + 2 * h);
    const float* pbp = pb + (size_t)(row0 + m) * S + (c0 + 2 * h);

    float2 dv[NT][4];
    float rowpart = 0.0f;
    #pragma unroll
    for (int t = 0; t < NT; ++t) {
        #pragma unroll
        for (int k = 0; k < 4; ++k) {
            float2 fa = *(const float2*)(pfp + 16 * t + 4 * k);
            float2 fb = *(const float2*)(pbp + 16 * t + 4 * k);
            float2 dd;
            dd.x = fa.x - fb.x;
            dd.y = fa.y - fb.y;
            dv[t][k] = dd;
            rowpart += dd.x + dd.y;
        }
    }

    // lanes m and m+16 hold complementary column sets of row m
    float tot = rowpart + __shfl_xor(rowpart, 16, 32);
    if (lane < 16) totalsLDS[w][m] = tot;
    __syncthreads();

    // ------ carry[r] = (logZ - logR[r]) + totals of chunks to the right ----
    if (lane < 16) {
        float c = logZ[0] - logR[row0 + lane];
        for (int wp = w + 1; wp < WAVES_PER_WG; ++wp) c += totalsLDS[wp][lane];
        carryLDS[w][lane] = c;
    }
    __syncthreads();

    // ------ WMMA suffix scan, right to left, all operands in registers -----
    float rc[8];                           // rc[j] = carry for row (j + 8h)
    #pragma unroll
    for (int j = 0; j < 8; ++j) rc[j] = carryLDS[w][j + 8 * h];

    // B fragments: U[k', n] = (k' >= n); v2f elem 0 -> K=4k+2h, elem 1 -> +1
    v2f bfrag[4];
    #pragma unroll
    for (int k = 0; k < 4; ++k) {
        int kr = 4 * k + 2 * h;
        bfrag[k].x = (kr     >= m) ? 1.0f : 0.0f;
        bfrag[k].y = (kr + 1 >= m) ? 1.0f : 0.0f;
    }

    float wl  = __powf(LAMBDA, (float)(S - 1 - (c0 + CH - 16 + m)));
    float acc = 0.0f;
    const int srcLane = lane & 16;        // column-0 holder in this half-wave

    #pragma unroll
    for (int t = NT - 1; t >= 0; --t) {
        v8f c;
        #pragma unroll
        for (int j = 0; j < 8; ++j) c[j] = rc[j];

        #pragma unroll
        for (int k = 0; k < 4; ++k) {
            v2f a;
            a.x = dv[t][k].x;
            a.y = dv[t][k].y;
            // D = A_k x U_k + C (chained accumulate)
            c = __builtin_amdgcn_wmma_f32_16x16x4_f32(
                    false, a, false, bfrag[k], (short)0, c, false, false);
        }

        float sq = 0.0f;
        #pragma unroll
        for (int j = 0; j < 8; ++j) sq += c[j] * c[j];
        acc += wl * sq;

        // next carry = D column 0 (old carry + full tile sum)
        #pragma unroll
        for (int j = 0; j < 8; ++j) rc[j] = __shfl(c[j], srcLane, 32);

        wl *= LAMBDA_P16;
    }

    // ------ reduce loss partials -------------------------------------------
    #pragma unroll
    for (int off = 16; off; off >>= 1) acc += __shfl_xor(acc, off, 32);
    if (lane == 0) wsum[w] = acc;
    __syncthreads();
    if (tid == 0) {
        float s = 0.0f;
        for (int i = 0; i < WAVES_PER_WG; ++i) s += wsum[i];
        partials[blockIdx.x] = s;
    }
}

// ---------------------------------------------------------------------------
// Dynamic fallback (loads in the scan loop) for shapes with odd CH.
// ---------------------------------------------------------------------------
__global__ void __launch_bounds__(512)
stb_loss_main_dyn(const float* __restrict__ pf,
                  const float* __restrict__ pb,
                  const float* __restrict__ logR,
                  const float* __restrict__ logZ,
                  float* __restrict__ partials,
                  int S, int CH) {
    __shared__ float totalsLDS[WAVES_PER_WG][16];
    __shared__ float carryLDS[WAVES_PER_WG][16];
    __shared__ float wsum[WAVES_PER_WG];

    const int tid  = threadIdx.x;
    const int lane = tid & 31;
    const int w    = tid >> 5;
    const int m    = lane & 15;
    const int h    = (lane >> 4) & 1;
    const int row0 = blockIdx.x * 16;
    const int c0   = w * CH;

    for (int r = 0; r < 16; ++r) {
        const float* pfr = pf + (size_t)(row0 + r) * S + c0;
        const float* pbr = pb + (size_t)(row0 + r) * S + c0;
        float s = 0.0f;
        for (int i = lane; i < CH; i += 32) s += pfr[i] - pbr[i];
        #pragma unroll
        for (int off = 16; off; off >>= 1) s += __shfl_xor(s, off, 32);
        if (lane == 0) totalsLDS[w][r] = s;
    }
    __syncthreads();

    if (lane < 16) {
        float c = logZ[0] - logR[row0 + lane];
        for (int wp = w + 1; wp < WAVES_PER_WG; ++wp) c += totalsLDS[wp][lane];
        carryLDS[w][lane] = c;
    }
    __syncthreads();

    float rc[8];
    #pragma unroll
    for (int j = 0; j < 8; ++j) rc[j] = carryLDS[w][j + 8 * h];

    v2f bfrag[4];
    #pragma unroll
    for (int k = 0; k < 4; ++k) {
        int kr = 4 * k + 2 * h;
        bfrag[k].x = (kr     >= m) ? 1.0f : 0.0f;
        bfrag[k].y = (kr + 1 >= m) ? 1.0f : 0.0f;
    }

    const float* pfp = pf + (size_t)(row0 + m) * S + (c0 + CH - 16 + 2 * h);
    const float* pbp = pb + (size_t)(row0 + m) * S + (c0 + CH - 16 + 2 * h);

    float wl  = __powf(LAMBDA, (float)(S - 1 - (c0 + CH - 16 + m)));
    float acc = 0.0f;
    const int srcLane = lane & 16;

    for (int t = 0, nT = CH >> 4; t < nT; ++t) {
        v8f c;
        #pragma unroll
        for (int j = 0; j < 8; ++j) c[j] = rc[j];
        #pragma unroll
        for (int k = 0; k < 4; ++k) {
            float2 fa = *(const float2*)(pfp + 4 * k);
            float2 fb = *(const float2*)(pbp + 4 * k);
            v2f a;
            a.x = fa.x - fb.x;
            a.y = fa.y - fb.y;
            c = __builtin_amdgcn_wmma_f32_16x16x4_f32(
                    false, a, false, bfrag[k], (short)0, c, false, false);
        }
        float sq = 0.0f;
        #pragma unroll
        for (int j = 0; j < 8; ++j) sq += c[j] * c[j];
        acc += wl * sq;
        #pragma unroll
        for (int j = 0; j < 8; ++j) rc[j] = __shfl(c[j], srcLane, 32);
        wl *= LAMBDA_P16;
        pfp -= 16;
        pbp -= 16;
    }

    #pragma unroll
    for (int off = 16; off; off >>= 1) acc += __shfl_xor(acc, off, 32);
    if (lane == 0) wsum[w] = acc;
    __syncthreads();
    if (tid == 0) {
        float s = 0.0f;
        for (int i = 0; i < WAVES_PER_WG; ++i) s += wsum[i];
        partials[blockIdx.x] = s;
    }
}

__global__ void __launch_bounds__(256)
stb_loss_reduce(const float* __restrict__ partials, int n,
                float* __restrict__ out, float invBS) {
    __shared__ float sm[256];
    float s = 0.0f;
    for (int i = threadIdx.x; i < n; i += 256) s += partials[i];
    sm[threadIdx.x] = s;
    __syncthreads();
    #pragma unroll
    for (int off = 128; off; off >>= 1) {
        if (threadIdx.x < off) sm[threadIdx.x] += sm[threadIdx.x + off];
        __syncthreads();
    }
    if (threadIdx.x == 0) out[0] = sm[0] * invBS;
}

extern "C" void kernel_launch(void* const* d_in, const int* in_sizes, int n_in,
                              void* d_out, int out_size, void* d_ws, size_t ws_size,
                              hipStream_t stream) {
    const float* pf   = (const float*)d_in[0];
    const float* pb   = (const float*)d_in[1];
    const float* logR = (const float*)d_in[2];
    const float* logZ = (const float*)d_in[3];
    float* out        = (float*)d_out;
    float* partials   = (float*)d_ws;

    const int B   = in_sizes[2];         // log_rewards has B elements
    const int S   = in_sizes[0] / B;     // [B,S] flat
    const int CH  = S / WAVES_PER_WG;    // 256 for S=4096
    const int nWG = B / 16;              // 16 rows per workgroup

    if (CH == 256) {                     // S = 4096: register-resident path
        stb_loss_main<16><<<nWG, 32 * WAVES_PER_WG, 0, stream>>>(
            pf, pb, logR, logZ, partials, S);
    } else if (CH == 128) {              // S = 2048
        stb_loss_main<8><<<nWG, 32 * WAVES_PER_WG, 0, stream>>>(
            pf, pb, logR, logZ, partials, S);
    } else {                             // generic
        stb_loss_main_dyn<<<nWG, 32 * WAVES_PER_WG, 0, stream>>>(
            pf, pb, logR, logZ, partials, S, CH);
    }

    const float invBS = 1.0f / ((float)B * (float)S);
    stb_loss_reduce<<<1, 256, 0, stream>>>(partials, nWG, out, invBS);
}